// CIF_65343632441386
// MI455X (gfx1250) — compile-verified
//
#include <hip/hip_runtime.h>

#define FEAT 256
#define BB   16
#define TT   4096
#define KWIN 5
#define KFLAT (KWIN * FEAT)   // 1280

typedef float v2f __attribute__((ext_vector_type(2)));
typedef float v8f __attribute__((ext_vector_type(8)));

// ---------------------------------------------------------------------------
// Kernel 0: fold linear weights into conv weights.
//   vflat[kw*FEAT + i] = sum_o conv_w[o, i, kw] * lin_w[o]
//   vflat[KFLAT]       = sum_o conv_b[o] * lin_w[o] + lin_b
// conv_w flat layout: o*KFLAT + i*KWIN + kw   (shape [O, I, K])
// ---------------------------------------------------------------------------
__global__ void cif_fold_kernel(const float* __restrict__ conv_w,
                                const float* __restrict__ conv_b,
                                const float* __restrict__ lin_w,
                                const float* __restrict__ lin_b,
                                float* __restrict__ vflat) {
    int j = blockIdx.x * blockDim.x + threadIdx.x;
    if (j < KFLAT) {
        int kw = j >> 8;            // j / FEAT
        int i  = j & (FEAT - 1);    // j % FEAT
        float s = 0.f;
        #pragma unroll 4
        for (int o = 0; o < FEAT; ++o)
            s = fmaf(conv_w[o * KFLAT + i * KWIN + kw], lin_w[o], s);
        vflat[j] = s;
    } else if (j == KFLAT) {
        float s = lin_b[0];
        for (int o = 0; o < FEAT; ++o)
            s = fmaf(conv_b[o], lin_w[o], s);
        vflat[KFLAT] = s;
    }
}

// ---------------------------------------------------------------------------
// Kernel 1: alphas[b, t] = sigmoid( xwin(b,t,:) . vflat + bias )
// One wave32 per 16-timestep tile; the 16x1280 dot runs through
// V_WMMA_F32_16X16X4_F32 (A = x window tile, B = vflat broadcast columns).
// A loads: one base pointer per conv tap, constant immediate offsets inside.
// B loads: vflat staged in LDS (shared by the block's 8 waves).
// ---------------------------------------------------------------------------
__global__ void cif_alpha_kernel(const float* __restrict__ x,
                                 const float* __restrict__ vflat,
                                 float* __restrict__ alphas) {
    __shared__ float s_v[KFLAT];

    // Cooperative stage of the folded weight vector (5 KB).
    for (int j = threadIdx.x; j < KFLAT; j += blockDim.x)
        s_v[j] = vflat[j];
    __syncthreads();

    const int lane  = threadIdx.x & 31;
    const int wave  = threadIdx.x >> 5;
    const int tile  = blockIdx.x * (blockDim.x >> 5) + wave;
    const int tilesPerB = TT / 16;
    const int b     = tile / tilesPerB;
    const int t0    = (tile % tilesPerB) * 16;
    const int m     = lane & 15;          // A-matrix row for this lane
    const int khalf = (lane >> 4) << 1;   // K sub-offset: 0 (lanes 0-15) / 2 (16-31)
    const float* xb = x + (size_t)b * TT * FEAT;
    const float cbias = vflat[KFLAT];

#if defined(__gfx1250__) && __has_builtin(__builtin_amdgcn_wmma_f32_16x16x4_f32)
    v8f c = {0.f, 0.f, 0.f, 0.f, 0.f, 0.f, 0.f, 0.f};

    // Wave-uniform edge test: only t0==0 and t0==TT-16 touch the padding.
    if (t0 >= 2 && t0 <= TT - 18) {
        // Interior: per lane, A elements (kk, kk+1) are adjacent channels of
        // one timestep. One b64 load per WMMA, immediate offsets carry i*4.
        const float* rowp = xb + (size_t)(t0 + m - 2) * FEAT + khalf;
        #pragma unroll
        for (int kw = 0; kw < KWIN; ++kw) {
            const float* rp = rowp + kw * FEAT;          // fixed base per tap
            const float* vp = s_v + kw * FEAT + khalf;   // fixed LDS base per tap
            #pragma unroll 8
            for (int i = 0; i < FEAT; i += 4) {
                v2f a  = *(const v2f*)(rp + i);
                v2f bv = *(const v2f*)(vp + i);
                c = __builtin_amdgcn_wmma_f32_16x16x4_f32(
                        false, a, false, bv, (short)0, c, false, false);
            }
        }
    } else {
        // Edge tiles: clamp the row address (hoisted per tap) so loads are
        // unconditional, then select 0 — cndmask only, no EXEC divergence.
        #pragma unroll
        for (int kw = 0; kw < KWIN; ++kw) {
            const int tt  = t0 + m + kw - 2;
            const int ttc = tt < 0 ? 0 : (tt > TT - 1 ? TT - 1 : tt);
            const bool ok = (tt == ttc);
            const float* rp = xb + (size_t)ttc * FEAT + khalf;
            const float* vp = s_v + kw * FEAT + khalf;
            #pragma unroll 8
            for (int i = 0; i < FEAT; i += 4) {
                v2f a  = *(const v2f*)(rp + i);
                v2f z  = {0.f, 0.f};
                a = ok ? a : z;
                v2f bv = *(const v2f*)(vp + i);
                c = __builtin_amdgcn_wmma_f32_16x16x4_f32(
                        false, a, false, bv, (short)0, c, false, false);
            }
        }
    }

    // D 16x16 f32 layout: VGPR r -> M=r (lanes 0-15, N=lane) / M=r+8 (lanes 16-31).
    // Column N=0 lives in lane 0 (M=0..7) and lane 16 (M=8..15).
    if (lane == 0 || lane == 16) {
        const int mbase = (lane >> 4) * 8;
        #pragma unroll
        for (int r = 0; r < 8; ++r) {
            float z = c[r] + cbias;
            alphas[(size_t)b * TT + t0 + mbase + r] = 1.f / (1.f + __expf(-z));
        }
    }
#else
    // Fallback (no f32 WMMA builtin): lanes 0-15 each compute one logit.
    if (lane < 16) {
        float s = cbias;
        for (int kk = 0; kk < KFLAT; ++kk) {
            int kw  = kk >> 8;
            int i   = kk & (FEAT - 1);
            int tt  = t0 + m + kw - 2;
            int ttc = tt < 0 ? 0 : (tt > TT - 1 ? TT - 1 : tt);
            float v = xb[(size_t)ttc * FEAT + i];
            s = fmaf((tt == ttc) ? v : 0.f, s_v[kk], s);
        }
        alphas[(size_t)b * TT + t0 + m] = 1.f / (1.f + __expf(-s));
    }
#endif
}

// ---------------------------------------------------------------------------
// Kernel 2: CIF integrate-and-fire. One block per batch. Alphas are staged
// coalesced into LDS; thread 0 runs the scalar scan (exact reference FP
// order) out of LDS, producing per-step segment weights + fire bits; then
// 256 threads (one per feature d) stream h coalesced and emit segments.
// ---------------------------------------------------------------------------
__global__ void cif_scan_kernel(const float* __restrict__ x,
                                const float* __restrict__ alphas,
                                float* __restrict__ out,   // [B, T, FEAT], pre-zeroed
                                int* __restrict__ lens) {  // [B]
    __shared__ float    wmain[TT];
    __shared__ float    wnext[TT];
    __shared__ unsigned fireb[TT / 32];
    __shared__ int      s_cnt;

    const int b = blockIdx.x;
    const int d = threadIdx.x;           // 0..255

    // Stage alphas into wmain (coalesced); thread 0 will overwrite in place.
    {
        const float* ab = alphas + (size_t)b * TT;
        for (int t = d; t < TT; t += FEAT)
            wmain[t] = ab[t];
        if (d < TT / 32) fireb[d] = 0u;
    }
    __syncthreads();

    if (threadIdx.x == 0) {
        float acc = 0.f;
        int cnt = 0;
        for (int t = 0; t < TT; ++t) {
            float a    = wmain[t];              // staged alpha
            float accn = acc + a;
            if (accn >= 1.0f) {                 // fire (beta == 1)
                float a1 = 1.0f - acc;          // mass completing this segment
                float a2 = a - a1;              // leftover into next segment
                wmain[t] = a1;
                wnext[t] = a2;
                fireb[t >> 5] |= (1u << (t & 31));
                acc = a2;
                ++cnt;
            } else {
                wmain[t] = a;
                wnext[t] = 0.f;
                acc = accn;
            }
        }
        s_cnt = cnt;
    }
    __syncthreads();

    const int cnt = s_cnt;
    const float* xb = x   + (size_t)b * TT * FEAT + d;
    float*       ob = out + (size_t)b * TT * FEAT + d;

    float s = 0.f;
    int seg = 0;
    for (int t = 0; t < TT; ++t) {
        float hu = xb[(size_t)t * FEAT];
        s = fmaf(wmain[t], hu, s);
        if (fireb[t >> 5] & (1u << (t & 31))) {
            ob[(size_t)seg * FEAT] = s;          // emitted segment vector
            s = wnext[t] * hu;                   // start next segment
            ++seg;
        }
    }
    if (cnt == 0) ob[0] = s;                     // no fire: emit final accumulator
    if (threadIdx.x == 0) lens[b] = (cnt > 0) ? cnt : 1;
}

// ---------------------------------------------------------------------------
extern "C" void kernel_launch(void* const* d_in, const int* in_sizes, int n_in,
                              void* d_out, int out_size, void* d_ws, size_t ws_size,
                              hipStream_t stream) {
    (void)in_sizes; (void)n_in; (void)out_size; (void)ws_size;

    const float* x      = (const float*)d_in[0];  // [B, T, FEAT]
    const float* conv_w = (const float*)d_in[1];  // [FEAT, FEAT, 5]
    const float* conv_b = (const float*)d_in[2];  // [FEAT]
    const float* lin_w  = (const float*)d_in[3];  // [FEAT, 1]
    const float* lin_b  = (const float*)d_in[4];  // [1]

    float* out    = (float*)d_out;                        // [B,T,FEAT] then [B] lens
    float* vflat  = (float*)d_ws;                         // KFLAT+1 floats
    float* alphas = vflat + (KFLAT + 32);                 // [B, T] floats
    int*   lens   = (int*)(out + (size_t)BB * TT * FEAT);

    // Non-fired output rows must be zero (harness poisons d_out).
    hipMemsetAsync(d_out, 0, (size_t)BB * TT * FEAT * sizeof(float), stream);

    cif_fold_kernel<<<(KFLAT + 1 + 255) / 256, 256, 0, stream>>>(
        conv_w, conv_b, lin_w, lin_b, vflat);

    const int tiles = BB * TT / 16;          // 4096 wave-tiles
    const int wavesPerBlock = 8;             // 256 threads = 8 wave32
    cif_alpha_kernel<<<tiles / wavesPerBlock, 32 * wavesPerBlock, 0, stream>>>(
        x, vflat, alphas);

    cif_scan_kernel<<<BB, FEAT, 0, stream>>>(x, alphas, out, lens);
}